// GwcCorrelation_12137577579280
// MI455X (gfx1250) — compile-verified
//
#include <hip/hip_runtime.h>
#include <hip/hip_bf16.h>

// CDNA5 / gfx1250: wave32, WMMA f32 16x16x4 for fp32-exact implicit-GEMM convs.
// NHWC intermediate layouts, LDS-staged pre-swizzled A fragments, branch-free
// halo masking (clamp + 0/1 multiply), 4-way N register blocking.
typedef float v2f __attribute__((ext_vector_type(2)));
typedef float v4f __attribute__((ext_vector_type(4)));
typedef float v8f __attribute__((ext_vector_type(8)));

#define BN_EPS 1e-5f

__device__ __forceinline__ float lrelu(float x) { return x >= 0.f ? x : 0.01f * x; }
__device__ __forceinline__ int clampi(int x, int lo, int hi) {
    return x < lo ? lo : (x > hi ? hi : x);
}

// ---------------------------------------------------------------------------
// 1) Group-wise correlation. Reads feat NCHW [2,128,64,96]; writes the
//    conv1 input directly in NHWC [162][64][96][16] using the reference's
//    row-major reshape map: flat=(b*16+g)*81+d2 -> nIdx=flat/16, c=flat%16.
//    Bandwidth-bound, K=8 contraction: VALU kernel.
// ---------------------------------------------------------------------------
__global__ __launch_bounds__(256)
void corr_kernel(const float* __restrict__ f1, const float* __restrict__ f2,
                 float* __restrict__ xout)
{
    const int HW = 64 * 96;
    int idx = blockIdx.x * blockDim.x + threadIdx.x;   // (b,g,h,w)
    int w = idx % 96;
    int t = idx / 96;
    int h = t % 64; t /= 64;
    int g = t % 16;
    int b = t / 16;
    if (b >= 2) return;

    float a[8];
    const float* f1p = f1 + ((long)b * 128 + g * 8) * HW + h * 96 + w;
#pragma unroll
    for (int c = 0; c < 8; ++c) a[c] = f1p[c * HW];

    const float* f2b = f2 + ((long)b * 128 + g * 8) * HW;
    const int flatBase = (b * 16 + g) * 81;            // + d2 -> (nIdx, c)

    for (int dy = 0; dy < 9; ++dy) {
        int h2 = h + dy - 4;
        for (int dx = 0; dx < 9; ++dx) {
            int w2 = w + dx - 4;
            float s = 0.f;
            if ((h2 >= 0) & (h2 < 64) & (w2 >= 0) & (w2 < 96)) {
                const float* p2 = f2b + h2 * 96 + w2;
#pragma unroll
                for (int c = 0; c < 8; ++c) s += a[c] * p2[c * HW];
            }
            int flat = flatBase + dy * 9 + dx;
            int nIdx = flat >> 4;
            int c    = flat & 15;
            xout[(((long)nIdx * 64 + h) * 96 + w) * 16 + c] = s * 0.125f;
        }
    }
}

// ---------------------------------------------------------------------------
// 2) Fused 3x3 conv + BN + leakyReLU via V_WMMA_F32_16X16X4_F32. NHWC in/out.
//    Block = one 16-cout tile x 512 pixels (8 waves x 64 pixels).
//    Weights staged in LDS pre-swizzled into A-fragment order [kstep][lane][2]
//    -> inner A fetch = 1 ds_load_b64 with immediate offset.
//    B fetch = 1 global_load_b64 (contiguous channels in NHWC) * 0/1 mask
//    (clamped address) -> no EXEC-masked control flow in the hot loop.
//    A frag (16x4 f32): lanes 0-15 -> K=0,1 ; lanes 16-31 -> K=2,3.
//    D (16x16 f32): VGPR r -> M=r (lanes 0-15), M=r+8 (lanes 16-31).
//    Grids exact; all waves reach __syncthreads/WMMA with EXEC all-ones.
// ---------------------------------------------------------------------------
template<int CIN, int STRIDE>
__global__ __launch_bounds__(256)
void conv3x3_bn_lrelu_wmma(const float* __restrict__ in, float* __restrict__ out,
                           const float* __restrict__ wgt,
                           const float* __restrict__ gam, const float* __restrict__ bet,
                           const float* __restrict__ mu,  const float* __restrict__ var,
                           int Hi, int Wi, int Cout, int Ho, int Wo)
{
    constexpr int nK = (CIN * 9) / 4;                  // K-steps of 4
    __shared__ __align__(16) float wsm[nK * 64];       // CIN=64 -> 36 KB
    const int nCoT   = Cout >> 4;
    const int coTile = blockIdx.x % nCoT;
    const int pixBlk = blockIdx.x / nCoT;

    // Cooperative swizzle: wsm[(s*32+ln)*2+j] = A[M=ln&15, K=4s+2*(ln>>4)+j]
    for (int i = threadIdx.x; i < nK * 64; i += 256) {
        int s  = i >> 6;
        int r  = i & 63;
        int ln = r >> 1;
        int j  = r & 1;
        int k   = s * 4 + (ln >> 4) * 2 + j;           // K = tap*CIN + ci
        int tap = k / CIN;
        int ci  = k - tap * CIN;
        wsm[i] = wgt[((coTile * 16 + (ln & 15)) * CIN + ci) * 9 + tap]; // OIHW
    }
    __syncthreads();

    const int lane  = threadIdx.x & 31;
    const int lan16 = lane & 15;
    const int half  = lane >> 4;
    const int waveId = threadIdx.x >> 5;
    const int pbase  = pixBlk * 512 + waveId * 64;

    int pn[4], pho[4], pwo[4];
#pragma unroll
    for (int t = 0; t < 4; ++t) {
        int p = pbase + t * 16 + lan16;
        pwo[t] = p % Wo;
        int q  = p / Wo;
        pho[t] = q % Ho;
        pn[t]  = q / Ho;
    }

    v8f acc[4] = {};
    const v2f* wa = (const v2f*)wsm + lane;            // + s*32 immediate below
    for (int tap = 0; tap < 9; ++tap) {
        const int ky = tap / 3;
        const int kx = tap - ky * 3;
        const float* bp[4];
        v2f mv[4];
#pragma unroll
        for (int t = 0; t < 4; ++t) {
            int hi = pho[t] * STRIDE + ky - 1;
            int wi = pwo[t] * STRIDE + kx - 1;
            float m = ((hi >= 0) & (hi < Hi) & (wi >= 0) & (wi < Wi)) ? 1.0f : 0.0f;
            int hic = clampi(hi, 0, Hi - 1);
            int wic = clampi(wi, 0, Wi - 1);
            bp[t] = in + ((long)(pn[t] * Hi + hic) * Wi + wic) * CIN;   // NHWC
            mv[t].x = m; mv[t].y = m;
        }
#pragma unroll 4
        for (int ci = 0; ci < CIN; ci += 4) {
            const int s  = tap * (CIN / 4) + ci / 4;
            const v2f a  = wa[s * 32];                 // ds_load_b64
            const int cA = ci + half * 2;
#pragma unroll
            for (int t = 0; t < 4; ++t) {
                v2f bb = *(const v2f*)(bp[t] + cA) * mv[t];   // global_load_b64
                acc[t] = __builtin_amdgcn_wmma_f32_16x16x4_f32(
                    false, a, false, bb, (short)0, acc[t], false, false);
            }
        }
    }

    float sc[8], sh[8];
#pragma unroll
    for (int r = 0; r < 8; ++r) {
        int co = coTile * 16 + half * 8 + r;
        sc[r] = gam[co] * rsqrtf(var[co] + BN_EPS);
        sh[r] = bet[co] - mu[co] * sc[r];
    }
#pragma unroll
    for (int t = 0; t < 4; ++t) {
        float* ob = out + ((long)(pn[t] * Ho + pho[t]) * Wo + pwo[t]) * Cout
                        + coTile * 16 + half * 8;      // NHWC, 32B aligned
        v4f lo, hi2;
#pragma unroll
        for (int r = 0; r < 4; ++r) lo[r]  = lrelu(acc[t][r]     * sc[r]     + sh[r]);
#pragma unroll
        for (int r = 0; r < 4; ++r) hi2[r] = lrelu(acc[t][r + 4] * sc[r + 4] + sh[r + 4]);
        *(v4f*)ob       = lo;                          // global_store_b128
        *(v4f*)(ob + 4) = hi2;
    }
}

// ---------------------------------------------------------------------------
// 3) ConvTranspose2d(32->16, k4, s2, p1) + BN + leakyReLU via WMMA. NHWC.
//    Output pixel parity fixes the 2x2 valid taps; N-tile = 16 same-parity
//    pixels in one row so (ky,kx) is wave-uniform. w5 staged in LDS as 4
//    parity-specialized A tables (32 KB).
// ---------------------------------------------------------------------------
__global__ __launch_bounds__(256)
void convT4x4s2_bn_lrelu_wmma(const float* __restrict__ in, float* __restrict__ out,
                              const float* __restrict__ w5,
                              const float* __restrict__ gam, const float* __restrict__ bet,
                              const float* __restrict__ mu,  const float* __restrict__ var)
{
    __shared__ __align__(16) float wsm[4 * 2048];      // 32 KB, [parity][s][lane][2]
    for (int i = threadIdx.x; i < 8192; i += 256) {
        int pc  = i >> 11;
        int rem = i & 2047;
        int s  = rem >> 6;
        int r  = rem & 63;
        int ln = r >> 1;
        int j  = r & 1;
        int k  = s * 4 + (ln >> 4) * 2 + j;            // K = tk*32 + ci
        int tk = k >> 5;
        int ci = k & 31;
        int ky = (pc >> 1) + 2 * (tk >> 1);
        int kx = (pc &  1) + 2 * (tk &  1);
        wsm[i] = w5[((ci * 16 + (ln & 15)) * 4 + ky) * 4 + kx];  // w5 IOHW (32,16,4,4)
    }
    __syncthreads();

    const int Hi = 32, Wi = 48, Ho = 64, Wo = 96;      // Cin=32, Cout=16
    const int lane  = threadIdx.x & 31;
    const int wave  = blockIdx.x * 8 + (threadIdx.x >> 5);
    const int rowIdx = wave / 6;                       // 6 tiles per output row
    const int tsub   = wave - rowIdx * 6;
    const int y = rowIdx % Ho;
    const int n = rowIdx / Ho;                         // grid exact: n < 162

    const int px    = tsub & 1;
    const int xb    = (tsub >> 1) * 16;
    const int half  = lane >> 4;
    const int lan16 = lane & 15;
    const int x = 2 * (xb + lan16) + px;
    const int kyPar = (y + 1) & 1;
    const int kxPar = (x + 1) & 1;                     // uniform across wave
    const v2f* wa = (const v2f*)wsm + (kyPar * 2 + kxPar) * 1024 + lane;

    v8f acc = {};
#pragma unroll
    for (int tk = 0; tk < 4; ++tk) {
        const int ky = kyPar + 2 * (tk >> 1);
        const int kx = kxPar + 2 * (tk & 1);
        const int yi = (y + 1 - ky) >> 1;
        const int xi = (x + 1 - kx) >> 1;
        float m = ((yi >= 0) & (yi < Hi) & (xi >= 0) & (xi < Wi)) ? 1.0f : 0.0f;
        int yic = clampi(yi, 0, Hi - 1);
        int xic = clampi(xi, 0, Wi - 1);
        const float* bp = in + ((long)(n * Hi + yic) * Wi + xic) * 32;   // NHWC
        v2f mv; mv.x = m; mv.y = m;
#pragma unroll 4
        for (int ci = 0; ci < 32; ci += 4) {
            const v2f a  = wa[(tk * 8 + (ci >> 2)) * 32];  // ds_load_b64
            const int cA = ci + half * 2;
            v2f bb = *(const v2f*)(bp + cA) * mv;          // global_load_b64
            acc = __builtin_amdgcn_wmma_f32_16x16x4_f32(
                false, a, false, bb, (short)0, acc, false, false);
        }
    }

    float* ob = out + (((long)n * Ho + y) * Wo + x) * 16 + half * 8;
    v4f lo, hi2;
#pragma unroll
    for (int r = 0; r < 4; ++r) {
        int co = half * 8 + r;
        float sc = gam[co] * rsqrtf(var[co] + BN_EPS);
        lo[r] = lrelu(acc[r] * sc + (bet[co] - mu[co] * sc));
    }
#pragma unroll
    for (int r = 0; r < 4; ++r) {
        int co = half * 8 + r + 4;
        float sc = gam[co] * rsqrtf(var[co] + BN_EPS);
        hi2[r] = lrelu(acc[r + 4] * sc + (bet[co] - mu[co] * sc));
    }
    *(v4f*)ob       = lo;                              // global_store_b128
    *(v4f*)(ob + 4) = hi2;
}

// ---------------------------------------------------------------------------
// 4) Final 3x3 conv, Cout=1 (+bias). NHWC input: 16 contiguous channels per
//    tap -> b128 loads. No matrix shape fits M=1: VALU kernel.
// ---------------------------------------------------------------------------
__global__ __launch_bounds__(256)
void conv_final_kernel(const float* __restrict__ in, float* __restrict__ out,
                       const float* __restrict__ w6, const float* __restrict__ bias6,
                       int N)
{
    const int H = 64, W = 96;
    int idx = blockIdx.x * blockDim.x + threadIdx.x;
    int x = idx % W;
    int t = idx / W;
    int y = t % H;
    int n = t / H;
    if (n >= N) return;
    float s = bias6[0];
#pragma unroll
    for (int ky = 0; ky < 3; ++ky) {
        int yy = y + ky - 1;
        if ((yy < 0) | (yy >= H)) continue;
#pragma unroll
        for (int kx = 0; kx < 3; ++kx) {
            int xx = x + kx - 1;
            if ((xx < 0) | (xx >= W)) continue;
            const float* ip = in + (((long)n * H + yy) * W + xx) * 16;   // NHWC
            int tap = ky * 3 + kx;
#pragma unroll
            for (int c = 0; c < 16; ++c) s += ip[c] * w6[c * 9 + tap];
        }
    }
    out[idx] = s;   // [162,1,64,96] flat == [2,81,64,96] flat
}

// ---------------------------------------------------------------------------
// Launch. All intermediates NHWC. Region A (64 MB): cv/act2/act4.
// Region B (128 MB): act1/act3/act5. Ping-pong, 192 MB workspace total.
// ---------------------------------------------------------------------------
extern "C" void kernel_launch(void* const* d_in, const int* in_sizes, int n_in,
                              void* d_out, int out_size, void* d_ws, size_t ws_size,
                              hipStream_t stream)
{
    const float* feat1 = (const float*)d_in[0];
    const float* feat2 = (const float*)d_in[1];
    const float* w1 = (const float*)d_in[2];
    const float* g1 = (const float*)d_in[3];
    const float* b1 = (const float*)d_in[4];
    const float* m1 = (const float*)d_in[5];
    const float* v1 = (const float*)d_in[6];
    const float* w2 = (const float*)d_in[7];
    const float* g2 = (const float*)d_in[8];
    const float* b2 = (const float*)d_in[9];
    const float* m2 = (const float*)d_in[10];
    const float* v2 = (const float*)d_in[11];
    const float* w3 = (const float*)d_in[12];
    const float* g3 = (const float*)d_in[13];
    const float* b3 = (const float*)d_in[14];
    const float* m3 = (const float*)d_in[15];
    const float* v3 = (const float*)d_in[16];
    const float* w4 = (const float*)d_in[17];
    const float* g4 = (const float*)d_in[18];
    const float* b4 = (const float*)d_in[19];
    const float* m4 = (const float*)d_in[20];
    const float* v4 = (const float*)d_in[21];
    const float* w5 = (const float*)d_in[22];
    const float* g5 = (const float*)d_in[23];
    const float* b5 = (const float*)d_in[24];
    const float* m5 = (const float*)d_in[25];
    const float* v5 = (const float*)d_in[26];
    const float* w6 = (const float*)d_in[27];
    const float* bias6 = (const float*)d_in[28];

    float* bufA = (float*)d_ws;                                        // 64 MB region
    float* bufB = (float*)((char*)d_ws + (size_t)64 * 1024 * 1024);    // 128 MB region

    const int Nimg = 162;  // B*81

    // corr -> NHWC [162,64,96,16]; 2*16*64*96 = 196608 threads
    corr_kernel<<<768, 256, 0, stream>>>(feat1, feat2, bufA);

    // conv1: NHWC [162,64,96,16] -> [162,64,96,32]; 1944 pixel blocks * 2 co tiles
    conv3x3_bn_lrelu_wmma<16, 1><<<3888, 256, 0, stream>>>(bufA, bufB, w1,
        g1, b1, m1, v1, 64, 96, 32, 64, 96);
    // conv2 (s=2): [162,64,96,32] -> [162,32,48,64]; 486 * 4
    conv3x3_bn_lrelu_wmma<32, 2><<<1944, 256, 0, stream>>>(bufB, bufA, w2,
        g2, b2, m2, v2, 64, 96, 64, 32, 48);
    // conv3: [162,32,48,64] -> [162,32,48,64]; 486 * 4
    conv3x3_bn_lrelu_wmma<64, 1><<<1944, 256, 0, stream>>>(bufA, bufB, w3,
        g3, b3, m3, v3, 32, 48, 64, 32, 48);
    // conv4: [162,32,48,64] -> [162,32,48,32]; 486 * 2
    conv3x3_bn_lrelu_wmma<64, 1><<<972, 256, 0, stream>>>(bufB, bufA, w4,
        g4, b4, m4, v4, 32, 48, 32, 32, 48);
    // convT: [162,32,48,32] -> [162,64,96,16]; 162*64 rows * 6 tiles = 62208 waves
    convT4x4s2_bn_lrelu_wmma<<<7776, 256, 0, stream>>>(bufA, bufB, w5, g5, b5, m5, v5);
    // conv6: [162,64,96,16] -> d_out [2,81,64,96]; 995328 threads
    conv_final_kernel<<<3888, 256, 0, stream>>>(bufB, (float*)d_out, w6, bias6, Nimg);

    (void)in_sizes; (void)n_in; (void)out_size; (void)ws_size;
}